// SphericalHarmonics_11347303596045
// MI455X (gfx1250) — compile-verified
//
#include <hip/hip_runtime.h>
#include <hip/hip_bf16.h>
#include <stdint.h>

// ---------------------------------------------------------------------------
// Real spherical harmonics, L<=20 (441 cols), 500k points, f32.
// Bandwidth-bound: 882 MB of output vs ~1.5 GFLOP of math -> design for the
// 23.3 TB/s HBM write stream. One point per lane, full-wave tile staged in
// LDS in destination layout, then streamed out with CDNA5 async
// global_store_async_from_lds_b128 (ASYNCcnt-tracked DMA, no VGPR staging).
// ---------------------------------------------------------------------------

#define L_MAX      20
#define LW         (L_MAX + 1)          // 21
#define K_COLS     441                  // (L+1)^2
#define TILE_PTS   32
#define TILE_FLOATS (TILE_PTS * K_COLS) // 14112
#define N_CHUNK128 (TILE_FLOATS / 128)  // 110 full-wave b128 chunks
#define TAIL_BASE  (N_CHUNK128 * 128)   // 14080 (32 floats remain)

// ----- compile-time tables (folded to literals after full unroll) ----------
constexpr double PI_D = 3.14159265358979323846264338327950288;

constexpr double csqrt(double x) {
  if (x <= 0.0) return 0.0;
  double g = x > 1.0 ? x : 1.0;
  for (int i = 0; i < 120; ++i) g = 0.5 * (g + x / g);
  return g;
}

struct Tbl { float v[LW * LW]; };

// N_{l,m} * (m>0 ? sqrt(2) : 1)
constexpr Tbl make_norm() {
  Tbl t{};
  for (int l = 0; l <= L_MAX; ++l)
    for (int m = 0; m <= l; ++m) {
      double ratio = 1.0;
      for (int k = l - m + 1; k <= l + m; ++k) ratio /= (double)k;
      double val = csqrt((2.0 * l + 1.0) / (4.0 * PI_D) * ratio);
      if (m > 0) val *= csqrt(2.0);
      t.v[l * LW + m] = (float)val;
    }
  return t;
}
// recurrence coefficients (2l-1)/(l-m) and (l+m-1)/(l-m)
constexpr Tbl make_ca() {
  Tbl t{};
  for (int m = 0; m <= L_MAX; ++m)
    for (int l = m + 2; l <= L_MAX; ++l)
      t.v[l * LW + m] = (float)((2.0 * l - 1.0) / (double)(l - m));
  return t;
}
constexpr Tbl make_cb() {
  Tbl t{};
  for (int m = 0; m <= L_MAX; ++m)
    for (int l = m + 2; l <= L_MAX; ++l)
      t.v[l * LW + m] = (float)((double)(l + m - 1) / (double)(l - m));
  return t;
}
constexpr Tbl NORM_T = make_norm();
constexpr Tbl CA_T   = make_ca();
constexpr Tbl CB_T   = make_cb();

// ---------------------------------------------------------------------------

__global__ void __launch_bounds__(32)
sh441_kernel(const float* __restrict__ coords, float* __restrict__ out, int npts) {
  __shared__ __align__(16) float tile[TILE_FLOATS];

  const int lane     = threadIdx.x;            // 0..31
  const int tileBase = blockIdx.x * TILE_PTS;  // first point of this tile
  int p  = tileBase + lane;
  int pc = p < npts ? p : npts - 1;            // clamp (tail safety)

  const float* cp = coords + (size_t)pc * 3;
  float x = cp[0], y = cp[1], z = cp[2];

  float r  = sqrtf(x * x + y * y + z * z);
  float ct = z / r;
  ct = fminf(1.0f, fmaxf(-1.0f, ct));
  float st = sqrtf(fmaxf(1.0f - ct * ct, 0.0f));

  // cos/sin of azimuth from x,y directly (== cos/sin(atan2(y,x)))
  float rho = sqrtf(x * x + y * y);
  float c1, s1;
  if (rho > 0.0f) { float ir = 1.0f / rho; c1 = x * ir; s1 = y * ir; }
  else            { c1 = 1.0f; s1 = 0.0f; }

  float* row = &tile[lane * K_COLS];   // lane-stride 441 words: odd -> no bank conflicts

  float pmm = 1.0f;                    // P_{m,m}, with Condon-Shortley phase
  float cm = 1.0f, sm = 0.0f;          // cos(m*az), sin(m*az)

#pragma unroll
  for (int m = 0; m <= L_MAX; ++m) {
    if (m > 0) {
      pmm = (float)(-(2 * m - 1)) * st * pmm;
      float cn = cm * c1 - sm * s1;
      float sn = sm * c1 + cm * s1;
      cm = cn; sm = sn;
    }
    { // l = m
      float b = NORM_T.v[m * LW + m] * pmm;
      int base = m * (m + 1);
      if (m == 0) row[base] = b;
      else { row[base + m] = b * cm; row[base - m] = b * sm; }
    }
    if (m < L_MAX) {
      float pprev = pmm;
      float pcur  = (float)(2 * m + 1) * ct * pmm;  // P_{m+1,m}
      {
        int l = m + 1, base = l * (l + 1);
        float b = NORM_T.v[l * LW + m] * pcur;
        if (m == 0) row[base] = b;
        else { row[base + m] = b * cm; row[base - m] = b * sm; }
      }
#pragma unroll
      for (int l = m + 2; l <= L_MAX; ++l) {
        float pn = CA_T.v[l * LW + m] * ct * pcur - CB_T.v[l * LW + m] * pprev;
        pprev = pcur; pcur = pn;
        float b = NORM_T.v[l * LW + m] * pn;
        int base = l * (l + 1);
        if (m == 0) row[base] = b;
        else { row[base + m] = b * cm; row[base - m] = b * sm; }
      }
    }
  }

  // LDS writes must land before the async DMA engine reads them.
  asm volatile("s_wait_dscnt 0x0" ::: "memory");

  int rows = npts - tileBase;
  if (rows > TILE_PTS) rows = TILE_PTS;

  uint32_t ldsBase = (uint32_t)(uintptr_t)&tile[0];   // low 32 bits = LDS offset
  float*   gBase   = out + (size_t)tileBase * K_COLS; // tile is contiguous in out

  if (rows == TILE_PTS) {
    // 110 full-wave 512B chunks, perfectly coalesced, no VGPR staging.
    uint64_t ga = (uint64_t)(uintptr_t)gBase + (uint32_t)(lane * 16);
    uint32_t la = ldsBase + (uint32_t)(lane * 16);
#pragma unroll 5
    for (int it = 0; it < N_CHUNK128; ++it) {
      asm volatile("global_store_async_from_lds_b128 %0, %1, off"
                   :: "v"(ga), "v"(la) : "memory");
      ga += 512u; la += 512u;
    }
    // tail: 32 remaining floats, one full-wave b32 async store
    uint64_t gt = (uint64_t)(uintptr_t)gBase + (uint32_t)(TAIL_BASE * 4 + lane * 4);
    uint32_t lt = ldsBase + (uint32_t)(TAIL_BASE * 4 + lane * 4);
    asm volatile("global_store_async_from_lds_b32 %0, %1, off"
                 :: "v"(gt), "v"(lt) : "memory");
  } else {
    // generic tail tile (not hit for 500000 points, 500000 % 32 == 0)
    int F = rows * K_COLS;
    for (int f = lane; f < F; f += 32) {
      uint64_t g = (uint64_t)(uintptr_t)gBase + (uint32_t)(f * 4);
      uint32_t l = ldsBase + (uint32_t)(f * 4);
      asm volatile("global_store_async_from_lds_b32 %0, %1, off"
                   :: "v"(g), "v"(l) : "memory");
    }
  }
  asm volatile("s_wait_asynccnt 0x0" ::: "memory");
}

// ---------------------------------------------------------------------------

extern "C" void kernel_launch(void* const* d_in, const int* in_sizes, int n_in,
                              void* d_out, int out_size, void* d_ws, size_t ws_size,
                              hipStream_t stream) {
  (void)n_in; (void)out_size; (void)d_ws; (void)ws_size;
  const float* coords = (const float*)d_in[0];
  float* out = (float*)d_out;
  int npts   = in_sizes[0] / 3;                 // coordinates: [N,3] f32
  int ntiles = (npts + TILE_PTS - 1) / TILE_PTS;
  sh441_kernel<<<dim3(ntiles), dim3(TILE_PTS), 0, stream>>>(coords, out, npts);
}